// GraphTransformerLayer_48799418417873
// MI455X (gfx1250) — compile-verified
//
#include <hip/hip_runtime.h>
#include <hip/hip_bf16.h>

#define B_   2
#define N_   512
#define D_   256
#define H_   8
#define HD_  32
#define DFF_ 1024

typedef __attribute__((ext_vector_type(2))) float v2f;
typedef __attribute__((ext_vector_type(8))) float v8f;

__device__ __forceinline__ v8f wmma_f32(v2f a, v2f b, v8f c) {
  // D = A(16x4) * B(4x16) + C, fp32 matrix core op (V_WMMA_F32_16X16X4_F32)
  return __builtin_amdgcn_wmma_f32_16x16x4_f32(false, a, false, b, (short)0, c,
                                               false, false);
}

// ---------------------------------------------------------------------------
// LayerNorm over last dim (D_=256), one row per block, 256 threads.
// ---------------------------------------------------------------------------
__global__ void ln256_kernel(const float* __restrict__ x,
                             const float* __restrict__ g,
                             const float* __restrict__ b,
                             float* __restrict__ y) {
  const size_t row = blockIdx.x;
  const int tid = threadIdx.x;
  float v = x[row * D_ + tid];
  float s = v, q = v * v;
#pragma unroll
  for (int off = 16; off; off >>= 1) {
    s += __shfl_xor(s, off, 32);
    q += __shfl_xor(q, off, 32);
  }
  __shared__ float s1[8], s2[8];
  const int lane = tid & 31, wid = tid >> 5;
  if (lane == 0) { s1[wid] = s; s2[wid] = q; }
  __syncthreads();
  float sa = 0.f, sq = 0.f;
#pragma unroll
  for (int i = 0; i < 8; ++i) { sa += s1[i]; sq += s2[i]; }
  const float m = sa * (1.0f / D_);
  const float var = sq * (1.0f / D_) - m * m;
  const float rs = rsqrtf(var + 1e-5f);
  y[row * D_ + tid] = (v - m) * rs * g[tid] + b[tid];
}

// ---------------------------------------------------------------------------
// Fused edge LayerNorm + Linear(D->H) + be, streamed once (512 MB read).
// Writes eb into scores buffer laid out (B,H,N,N).
// One block per (b,i); 8 waves, each wave owns one j at a time.
// We/gamma/beta for this lane's 8 d-slots are preloaded into registers so the
// inner loop's only global traffic is the nontemporal 1KB edge-row stream.
// ---------------------------------------------------------------------------
__global__ void edge_bias_kernel(const float* __restrict__ E,
                                 const float* __restrict__ g,
                                 const float* __restrict__ bb,
                                 const float* __restrict__ We,
                                 const float* __restrict__ be,
                                 float* __restrict__ scores) {
  const int bi = blockIdx.x;           // b*N + i
  const int b = bi / N_, i = bi % N_;
  const int wave = threadIdx.x >> 5, lane = threadIdx.x & 31;
  const float* Erow = E + (size_t)bi * N_ * D_;

  float Wr[8][8], gr[8], br[8];
#pragma unroll
  for (int r = 0; r < 8; ++r) {
    const int d = lane + 32 * r;
    gr[r] = g[d];
    br[r] = bb[d];
#pragma unroll
    for (int h = 0; h < H_; ++h) Wr[r][h] = We[d * H_ + h];
  }

  for (int j = wave; j < N_; j += 8) {
    const float* e = Erow + (size_t)j * D_;
    if (j + 8 < N_)
      __builtin_prefetch(Erow + (size_t)(j + 8) * D_ + lane * 8, 0, 1);
    float vals[8];
    float s = 0.f, q = 0.f;
#pragma unroll
    for (int r = 0; r < 8; ++r) {
      const float t = __builtin_nontemporal_load(&e[lane + 32 * r]);
      vals[r] = t; s += t; q += t * t;
    }
#pragma unroll
    for (int off = 16; off; off >>= 1) {
      s += __shfl_xor(s, off, 32);
      q += __shfl_xor(q, off, 32);
    }
    const float m = s * (1.0f / D_);
    const float rs = rsqrtf(q * (1.0f / D_) - m * m + 1e-5f);
    float acc[8] = {0.f, 0.f, 0.f, 0.f, 0.f, 0.f, 0.f, 0.f};
#pragma unroll
    for (int r = 0; r < 8; ++r) {
      const float nv = (vals[r] - m) * rs * gr[r] + br[r];
#pragma unroll
      for (int h = 0; h < H_; ++h) acc[h] += nv * Wr[r][h];
    }
#pragma unroll
    for (int h = 0; h < H_; ++h) {
#pragma unroll
      for (int off = 16; off; off >>= 1) acc[h] += __shfl_xor(acc[h], off, 32);
    }
    if (lane < H_) {
      float o = acc[0];
#pragma unroll
      for (int h = 1; h < H_; ++h) o = (lane == h) ? acc[h] : o;
      const size_t idx = ((((size_t)b * H_ + lane) * N_ + i) * N_ + j);
      scores[idx] = o + be[lane];
    }
  }
}

// ---------------------------------------------------------------------------
// Generic fp32 WMMA GEMM: C[M,N] = act(A[M,K] @ W[K,N] + bias) (+ residual).
// Block = 256 threads = 8 waves arranged 2(M) x 4(N); block tile 32x64.
// K processed in 128-wide LDS-staged chunks.
// ACT: 0 = none, 1 = exact GELU.  RES: 1 = add residual R.
// ---------------------------------------------------------------------------
template <int ACT, int RES>
__global__ void wmma_gemm_kernel(const float* __restrict__ A,
                                 const float* __restrict__ W,
                                 const float* __restrict__ bias,
                                 const float* __restrict__ R,
                                 float* __restrict__ C,
                                 int M, int N, int K) {
  __shared__ float As[32][132];   // +4 pad, 16B-aligned rows
  __shared__ float Ws[128][68];
  const int wave = threadIdx.x >> 5, lane = threadIdx.x & 31;
  const int half = lane >> 4, l = lane & 15;
  const int wm = wave >> 2, wn = wave & 3;
  const int m0 = blockIdx.y * 32, n0 = blockIdx.x * 64;

  v8f c = {};
  for (int kc = 0; kc < K; kc += 128) {
    __syncthreads();
    // Stage A chunk: 32 x 128 floats (1024 float4, 4 per thread)
    for (int t = threadIdx.x; t < 32 * 32; t += 256) {
      const int r = t >> 5, cq = t & 31;
      *(float4*)&As[r][cq * 4] =
          *(const float4*)&A[(size_t)(m0 + r) * K + kc + cq * 4];
    }
    // Stage W chunk: 128 x 64 floats (2048 float4, 8 per thread)
    for (int t = threadIdx.x; t < 128 * 16; t += 256) {
      const int r = t >> 4, cq = t & 15;
      *(float4*)&Ws[r][cq * 4] =
          *(const float4*)&W[(size_t)(kc + r) * N + n0 + cq * 4];
    }
    __syncthreads();
#pragma unroll
    for (int kk = 0; kk < 128; kk += 4) {
      v2f a, bf;
      a.x = As[wm * 16 + l][kk + 2 * half];
      a.y = As[wm * 16 + l][kk + 2 * half + 1];
      bf.x = Ws[kk + 2 * half][wn * 16 + l];
      bf.y = Ws[kk + 2 * half + 1][wn * 16 + l];
      c = wmma_f32(a, bf, c);
    }
  }
#pragma unroll
  for (int r = 0; r < 8; ++r) {
    const int row = m0 + wm * 16 + r + 8 * half;
    const int col = n0 + wn * 16 + l;
    float v = c[r] + bias[col];
    if (ACT == 1) v = 0.5f * v * (1.0f + erff(v * 0.70710678118654752f));
    if (RES) v += R[(size_t)row * N + col];
    C[(size_t)row * N + col] = v;
  }
}

// ---------------------------------------------------------------------------
// scores = q @ k^T / sqrt(HD) + eb (already in buffer), masked.
// q,k stored as (B,N,D) with head h at column h*HD. Grid (N/16, B*H).
// Block covers 16 rows x 512 cols; each wave does 4 16x16 tiles, K=32.
// ---------------------------------------------------------------------------
__global__ void scores_kernel(const float* __restrict__ qb,
                              const float* __restrict__ kb,
                              const int* __restrict__ mask,
                              float* __restrict__ scores) {
  const int bh = blockIdx.y;
  const int b = bh / H_, h = bh % H_;
  const int i0 = blockIdx.x * 16;
  const int wave = threadIdx.x >> 5, lane = threadIdx.x & 31;
  const int half = lane >> 4, l = lane & 15;
  const float* qbase = qb + (size_t)b * N_ * D_ + (size_t)h * HD_;
  const float* kbase = kb + (size_t)b * N_ * D_ + (size_t)h * HD_;

  __shared__ float qs[16][HD_ + 4];
  for (int t = threadIdx.x; t < 16 * HD_; t += 256) {
    const int r = t >> 5, cc = t & 31;
    qs[r][cc] = qbase[(size_t)(i0 + r) * D_ + cc];
  }
  __syncthreads();

  const float scale = 0.17677669529663687f;  // 1/sqrt(32)
#pragma unroll
  for (int jt = 0; jt < 4; ++jt) {
    const int j0 = (wave + jt * 8) * 16;
    v8f c = {};
#pragma unroll
    for (int kk = 0; kk < HD_; kk += 4) {
      v2f a, bf;
      a.x = qs[l][kk + 2 * half];
      a.y = qs[l][kk + 2 * half + 1];
      const float* kp = kbase + (size_t)(j0 + l) * D_ + kk + 2 * half;
      bf.x = kp[0];
      bf.y = kp[1];
      c = wmma_f32(a, bf, c);
    }
#pragma unroll
    for (int r = 0; r < 8; ++r) {
      const int i = i0 + r + 8 * half;
      const int j = j0 + l;
      const size_t idx = ((size_t)bh * N_ + i) * N_ + j;
      const float val = c[r] * scale + scores[idx];
      const int mk = mask[((size_t)b * N_ + i) * N_ + j];
      scores[idx] = (mk == 0) ? -__builtin_inff() : val;
    }
  }
}

// ---------------------------------------------------------------------------
// Row softmax over N_=512, in place. One block (256 thr) per (b,h,i) row.
// ---------------------------------------------------------------------------
__global__ void softmax_kernel(float* __restrict__ scores) {
  float* s = scores + (size_t)blockIdx.x * N_;
  const int tid = threadIdx.x, lane = tid & 31, wid = tid >> 5;
  const float v0 = s[tid], v1 = s[tid + 256];
  float mx = fmaxf(v0, v1);
#pragma unroll
  for (int off = 16; off; off >>= 1) mx = fmaxf(mx, __shfl_xor(mx, off, 32));
  __shared__ float sm[8], ss[8];
  if (lane == 0) sm[wid] = mx;
  __syncthreads();
  float m = sm[0];
#pragma unroll
  for (int i = 1; i < 8; ++i) m = fmaxf(m, sm[i]);
  const float e0 = __expf(v0 - m), e1 = __expf(v1 - m);
  float sum = e0 + e1;
#pragma unroll
  for (int off = 16; off; off >>= 1) sum += __shfl_xor(sum, off, 32);
  if (lane == 0) ss[wid] = sum;
  __syncthreads();
  float st = 0.f;
#pragma unroll
  for (int i = 0; i < 8; ++i) st += ss[i];
  const float inv = 1.0f / st;
  s[tid] = e0 * inv;
  s[tid + 256] = e1 * inv;
}

// ---------------------------------------------------------------------------
// out[b, i, h*HD+d] = sum_j attn[b,h,i,j] * v[b, j, h*HD+d].
// Grid (8, B*H); each wave owns one 16x16 output tile (mt 0..31, nt 0..1),
// K = 512 via 128 fp32 WMMA steps.
// ---------------------------------------------------------------------------
__global__ void attn_v_kernel(const float* __restrict__ P,
                              const float* __restrict__ vb,
                              float* __restrict__ out) {
  const int bh = blockIdx.y;
  const int b = bh / H_, h = bh % H_;
  const int wave = threadIdx.x >> 5, lane = threadIdx.x & 31;
  const int half = lane >> 4, l = lane & 15;
  const int t = blockIdx.x * 8 + wave;  // 0..63
  const int i0 = (t >> 1) * 16, n0 = (t & 1) * 16;
  const float* Pb = P + (size_t)bh * N_ * N_;
  const float* Vb = vb + (size_t)b * N_ * D_ + (size_t)h * HD_;

  v8f c = {};
  for (int kk = 0; kk < N_; kk += 4) {
    v2f a, bf;
    const float* pp = Pb + (size_t)(i0 + l) * N_ + kk + 2 * half;
    a.x = pp[0];
    a.y = pp[1];
    const float* vp = Vb + (size_t)(kk + 2 * half) * D_ + n0 + l;
    bf.x = vp[0];
    bf.y = vp[D_];
    c = wmma_f32(a, bf, c);
  }
#pragma unroll
  for (int r = 0; r < 8; ++r) {
    const int i = i0 + r + 8 * half;
    out[((size_t)b * N_ + i) * D_ + h * HD_ + n0 + l] = c[r];
  }
}

// ---------------------------------------------------------------------------
extern "C" void kernel_launch(void* const* d_in, const int* in_sizes, int n_in,
                              void* d_out, int out_size, void* d_ws,
                              size_t ws_size, hipStream_t stream) {
  const float* node   = (const float*)d_in[0];
  const float* edge   = (const float*)d_in[1];
  const int*   adj    = (const int*)d_in[2];
  const float* nng    = (const float*)d_in[3];
  const float* nnb    = (const float*)d_in[4];
  const float* eng    = (const float*)d_in[5];
  const float* enb    = (const float*)d_in[6];
  const float* Wq     = (const float*)d_in[7];
  const float* bq     = (const float*)d_in[8];
  const float* Wk     = (const float*)d_in[9];
  const float* bk     = (const float*)d_in[10];
  const float* Wv     = (const float*)d_in[11];
  const float* bv     = (const float*)d_in[12];
  const float* Wo     = (const float*)d_in[13];
  const float* bo     = (const float*)d_in[14];
  const float* We     = (const float*)d_in[15];
  const float* be     = (const float*)d_in[16];
  const float* ffg    = (const float*)d_in[17];
  const float* ffb    = (const float*)d_in[18];
  const float* W1     = (const float*)d_in[19];
  const float* b1     = (const float*)d_in[20];
  const float* W2     = (const float*)d_in[21];
  const float* b2     = (const float*)d_in[22];

  const size_t SZ = (size_t)B_ * N_ * D_;  // 262144
  float* ws   = (float*)d_ws;
  float* nn   = ws;            // LN(node)
  float* q    = ws + 1 * SZ;
  float* k    = ws + 2 * SZ;
  float* v    = ws + 3 * SZ;
  float* ao   = ws + 4 * SZ;   // attn output, (B,N,D)
  float* x    = ws + 5 * SZ;   // residual state after attention
  float* hln  = ws + 6 * SZ;   // LN(x)
  float* h1   = ws + 7 * SZ;   // (B*N, DFF) = 4*SZ floats
  float* sc   = ws + 11 * SZ;  // scores/attn (B,H,N,N) = 16*SZ... (4M floats)

  const dim3 blk(256);

  // 1. node LayerNorm
  ln256_kernel<<<dim3(B_ * N_), blk, 0, stream>>>(node, nng, nnb, nn);

  // 2. fused edge LN + We projection -> eb, streamed into scores buffer
  edge_bias_kernel<<<dim3(B_ * N_), blk, 0, stream>>>(edge, eng, enb, We, be, sc);

  // 3. Q, K, V projections (M=1024, N=256, K=256)
  const dim3 gqkv(D_ / 64, (B_ * N_) / 32);
  wmma_gemm_kernel<0, 0><<<gqkv, blk, 0, stream>>>(nn, Wq, bq, nullptr, q,
                                                   B_ * N_, D_, D_);
  wmma_gemm_kernel<0, 0><<<gqkv, blk, 0, stream>>>(nn, Wk, bk, nullptr, k,
                                                   B_ * N_, D_, D_);
  wmma_gemm_kernel<0, 0><<<gqkv, blk, 0, stream>>>(nn, Wv, bv, nullptr, v,
                                                   B_ * N_, D_, D_);

  // 4. scores = q k^T * scale + eb, mask
  scores_kernel<<<dim3(N_ / 16, B_ * H_), blk, 0, stream>>>(q, k, adj, sc);

  // 5. softmax
  softmax_kernel<<<dim3(B_ * H_ * N_), blk, 0, stream>>>(sc);

  // 6. attn @ V -> (B,N,D)
  attn_v_kernel<<<dim3(8, B_ * H_), blk, 0, stream>>>(sc, v, ao);

  // 7. output projection + residual: x = node + ao @ Wo + bo
  wmma_gemm_kernel<0, 1><<<gqkv, blk, 0, stream>>>(ao, Wo, bo, node, x,
                                                   B_ * N_, D_, D_);

  // 8. FFN LayerNorm
  ln256_kernel<<<dim3(B_ * N_), blk, 0, stream>>>(x, ffg, ffb, hln);

  // 9. h1 = gelu(hln @ W1 + b1)   (M=1024, N=1024, K=256)
  wmma_gemm_kernel<1, 0><<<dim3(DFF_ / 64, (B_ * N_) / 32), blk, 0, stream>>>(
      hln, W1, b1, nullptr, h1, B_ * N_, DFF_, D_);

  // 10. out = x + h1 @ W2 + b2    (M=1024, N=256, K=1024)
  wmma_gemm_kernel<0, 1><<<gqkv, blk, 0, stream>>>(h1, W2, b2, x,
                                                   (float*)d_out, B_ * N_, D_,
                                                   DFF_);
}